// UltraHopfieldLayer_20624432955867
// MI455X (gfx1250) — compile-verified
//
#include <hip/hip_runtime.h>
#include <hip/hip_bf16.h>

#define NN 8192
#define RPB 32      // rows per block in main kernel
#define TPB 256
#define NBLK 256    // main-kernel grid (NN / RPB)

typedef float v2f __attribute__((ext_vector_type(2)));
typedef float v4f __attribute__((ext_vector_type(4)));
typedef float v8f __attribute__((ext_vector_type(8)));

static __device__ __forceinline__ v4f ldnt(const float* p) {
    return __builtin_nontemporal_load(reinterpret_cast<const v4f*>(p));
}

// ---------------- kernel 0: d[i] = || coords[i] - coords[dest] || ----------------
__global__ __launch_bounds__(256) void k_dist(const float* __restrict__ coords,
                                              const int* __restrict__ dstp,
                                              float* __restrict__ wsd) {
    int i = blockIdx.x * 256 + threadIdx.x;
    int dst = *dstp;
    float cx = coords[2 * dst], cy = coords[2 * dst + 1];
    if (i < NN) {
        float dx = coords[2 * i] - cx;
        float dy = coords[2 * i + 1] - cy;
        wsd[i] = sqrtf(dx * dx + dy * dy);
    }
}

// ---------------- kernel 1: streaming elementwise pass + partial reductions ------
__global__ __launch_bounds__(TPB) void k_main(const float* __restrict__ logits,
                                              const float* __restrict__ dist,
                                              const float* __restrict__ wsd,
                                              float* __restrict__ outf,
                                              float* __restrict__ colpart,
                                              float* __restrict__ scalpart) {
    __shared__ float s_row[RPB][8];   // per-row, per-wave partial row sums
    __shared__ float s_scal[8][4];    // per-wave scalar partials

    const int t = threadIdx.x;
    const int lane = t & 31;
    const int wave = t >> 5;
    const int row0 = blockIdx.x * RPB;

    // Fixed column ownership: thread t owns float4 chunks at element 4*(t + 256*q)
    v4f dj[8], colacc[8];
    const v4f zero4 = {0.f, 0.f, 0.f, 0.f};
#pragma unroll
    for (int q = 0; q < 8; ++q) {
        dj[q] = *reinterpret_cast<const v4f*>(wsd + 4 * (t + TPB * q));
        colacc[q] = zero4;
    }

    float ne = 0.f, pc = 0.f, bp = 0.f, hh = 0.f;
    const float K = -2.0f * 1.44269504088896340736f;  // -2 * log2(e)

    for (int r = 0; r < RPB; ++r) {
        const int row = row0 + r;
        const float di = wsd[row];
        const float* lrow = logits + (size_t)row * NN;
        const float* drow = dist + (size_t)row * NN;

        // speculative prefetch of next row (gfx1250 global_prefetch_b8)
        if (row + 1 < NN) {
            __builtin_prefetch(lrow + NN + 4 * t, 0, 0);
            __builtin_prefetch(drow + NN + 4 * t, 0, 0);
        }

        float rowp = 0.f;
#pragma unroll
        for (int q = 0; q < 8; ++q) {
            const int cb = 4 * (t + TPB * q);
            v4f la = ldnt(lrow + cb);
            v4f da = ldnt(drow + cb);
#pragma unroll
            for (int c = 0; c < 4; ++c) {
                float L = la[c], D = da[c];
                bool valid = D < 1.0e6f;
                // sigmoid(L / 0.5) = 1 / (1 + exp(-2L)) = rcp(1 + exp2(-2L*log2e))
                float e = __builtin_amdgcn_exp2f(K * L);
                float sg = __builtin_amdgcn_rcpf(1.0f + e);
                float x = valid ? sg : 0.0f;
                ne += valid ? 1.0f : 0.0f;
                pc = fmaf(D, x, pc);        // invalid: D*0 == 0 (D = 1e9, finite)
                bp += x - x * x;            // x*(1-x)
                float df = di - dj[q][c];
                hh = fmaf(x, fmaxf(df, 0.0f), hh);
                rowp += x;
                colacc[q][c] += x;
            }
        }
        // wave-level row-sum reduction (no barriers in hot loop)
#pragma unroll
        for (int off = 16; off; off >>= 1) rowp += __shfl_xor(rowp, off, 32);
        if (lane == 0) s_row[r][wave] = rowp;
    }

    // wave-level scalar reductions
#pragma unroll
    for (int off = 16; off; off >>= 1) {
        ne += __shfl_xor(ne, off, 32);
        pc += __shfl_xor(pc, off, 32);
        bp += __shfl_xor(bp, off, 32);
        hh += __shfl_xor(hh, off, 32);
    }
    if (lane == 0) {
        s_scal[wave][0] = ne; s_scal[wave][1] = pc;
        s_scal[wave][2] = bp; s_scal[wave][3] = hh;
    }
    __syncthreads();

    if (t < RPB) {  // deterministic 8-way sum per row
        float s = 0.f;
#pragma unroll
        for (int w = 0; w < 8; ++w) s += s_row[t][w];
        outf[row0 + t] = s;
    }
    if (t == 0) {
        float a0 = 0.f, a1 = 0.f, a2 = 0.f, a3 = 0.f;
#pragma unroll
        for (int w = 0; w < 8; ++w) {
            a0 += s_scal[w][0]; a1 += s_scal[w][1];
            a2 += s_scal[w][2]; a3 += s_scal[w][3];
        }
        scalpart[blockIdx.x * 4 + 0] = a0;
        scalpart[blockIdx.x * 4 + 1] = a1;
        scalpart[blockIdx.x * 4 + 2] = a2;
        scalpart[blockIdx.x * 4 + 3] = a3;
    }

    // store per-block column partials (coalesced b128)
#pragma unroll
    for (int q = 0; q < 8; ++q) {
        *reinterpret_cast<v4f*>(colpart + (size_t)blockIdx.x * NN + 4 * (t + TPB * q)) =
            colacc[q];
    }
}

// ------- kernel 2: in_flow[j] = sum_b colpart[b][j], via WMMA with ones A -------
// C(16x16) += A(16x4, all ones) * B(4x16 tile of colpart). After looping over all
// 256 block-partials, every row of C holds the 16 column sums.
__global__ __launch_bounds__(256) void k_colreduce(const float* __restrict__ colpart,
                                                   float* __restrict__ inf) {
    const int lane = threadIdx.x & 31;
    const int wave = threadIdx.x >> 5;
    const int j0 = (blockIdx.x * 8 + wave) * 16;  // 16 columns per wave
    const int n = lane & 15;
    const int half = lane >> 4;

    v8f c = {0.f, 0.f, 0.f, 0.f, 0.f, 0.f, 0.f, 0.f};
    v2f a; a[0] = 1.0f; a[1] = 1.0f;  // A = ones(16x4) in WMMA layout

    for (int b = 0; b < NBLK; b += 4) {
        // B layout (4x16 f32): VGPR0 = {K=0 lanes0-15, K=2 lanes16-31},
        //                      VGPR1 = {K=1 lanes0-15, K=3 lanes16-31}
        int r0 = b + 2 * half;
        v2f bb;
        bb[0] = colpart[(size_t)r0 * NN + j0 + n];
        bb[1] = colpart[(size_t)(r0 + 1) * NN + j0 + n];
        c = __builtin_amdgcn_wmma_f32_16x16x4_f32(false, a, false, bb,
                                                  (short)0, c, false, false);
    }
    // C VGPR0: lanes 0-15 hold (M=0, N=lane) -> the column sums
    if (lane < 16) inf[j0 + lane] = c[0];
}

// ---------------- kernel 3: flow penalty + final energy (single block) ----------
__global__ __launch_bounds__(256) void k_final(const float* __restrict__ outf,
                                               const float* __restrict__ inf,
                                               const float* __restrict__ scalpart,
                                               const int* __restrict__ srcp,
                                               const int* __restrict__ dstp,
                                               float* __restrict__ out) {
    __shared__ double red[8][5];
    const int t = threadIdx.x;
    const int lane = t & 31;
    const int wave = t >> 5;
    const int src = *srcp, dst = *dstp;

    double fp = 0.0;
    for (int j = t; j < NN; j += 256) {
        float tgt = (j == src ? 1.0f : 0.0f) - (j == dst ? 1.0f : 0.0f);
        float v = outf[j] - inf[j] - tgt;
        fp += (double)v * (double)v;
    }
    double ne = scalpart[t * 4 + 0];
    double pc = scalpart[t * 4 + 1];
    double bp = scalpart[t * 4 + 2];
    double hh = scalpart[t * 4 + 3];

#pragma unroll
    for (int off = 16; off; off >>= 1) {
        fp += __shfl_xor(fp, off, 32);
        ne += __shfl_xor(ne, off, 32);
        pc += __shfl_xor(pc, off, 32);
        bp += __shfl_xor(bp, off, 32);
        hh += __shfl_xor(hh, off, 32);
    }
    if (lane == 0) {
        red[wave][0] = fp; red[wave][1] = ne; red[wave][2] = pc;
        red[wave][3] = bp; red[wave][4] = hh;
    }
    __syncthreads();
    if (t == 0) {
        double F = 0, E = 0, P = 0, B = 0, H = 0;
#pragma unroll
        for (int w = 0; w < 8; ++w) {
            F += red[w][0]; E += red[w][1]; P += red[w][2];
            B += red[w][3]; H += red[w][4];
        }
        const double n = (double)NN;
        const double nn = n * n;
        double density = E / nn;
        double mu2 = 10.0 * (1.0 + density);
        double mu3 = mu2;
        double heuristic = H / n;
        double energy = P / (E + 1e-6) + mu2 * F / n + mu3 * B / nn
                        - 0.5 * heuristic;
        out[0] = (float)energy;
    }
}

extern "C" void kernel_launch(void* const* d_in, const int* in_sizes, int n_in,
                              void* d_out, int out_size, void* d_ws, size_t ws_size,
                              hipStream_t stream) {
    const float* logits = (const float*)d_in[0];
    const float* dist   = (const float*)d_in[1];
    const float* coords = (const float*)d_in[2];
    const int*   srcp   = (const int*)d_in[3];
    const int*   dstp   = (const int*)d_in[4];

    float* ws      = (float*)d_ws;
    float* wsd     = ws;                 // 8192: d[i]
    float* outf    = ws + 8192;          // 8192: out_flow
    float* inf     = ws + 16384;         // 8192: in_flow
    float* scal    = ws + 24576;         // 256*4 scalar partials
    float* colpart = ws + 25600;         // 256*8192 column partials (8 MB)

    k_dist     <<<NN / 256, 256, 0, stream>>>(coords, dstp, wsd);
    k_main     <<<NBLK,     TPB, 0, stream>>>(logits, dist, wsd, outf, colpart, scal);
    k_colreduce<<<NN / 128, 256, 0, stream>>>(colpart, inf);
    k_final    <<<1,        256, 0, stream>>>(outf, inf, scal, srcp, dstp, (float*)d_out);
}